// Attention3D_78073915507409
// MI455X (gfx1250) — compile-verified
//
#include <hip/hip_runtime.h>

typedef float v2f __attribute__((ext_vector_type(2)));
typedef float v8f __attribute__((ext_vector_type(8)));

// ---------------------------------------------------------------------------
// Encoder conv: stride 2, zero-pad 1, 3x3x3, ReLU.
// All OC computed per thread: each input tap loaded once, used OC times.
// grid = (spatial/256, 1, B)
// ---------------------------------------------------------------------------
template <int IC, int OC>
__global__ void enc_conv_kernel(const float* __restrict__ in,
                                const float* __restrict__ w,
                                const float* __restrict__ bias,
                                float* __restrict__ out,
                                int inS, int outS) {
  const int b = blockIdx.z;
  const int n = outS * outS * outS;
  const int sp = blockIdx.x * blockDim.x + threadIdx.x;
  if (sp >= n) return;
  const int ow = sp % outS;
  int t = sp / outS;
  const int oh = t % outS, od = t / outS;
  const int inS2 = inS * inS;
  float s[OC];
#pragma unroll
  for (int oc = 0; oc < OC; ++oc) s[oc] = bias[oc];
  for (int ic = 0; ic < IC; ++ic) {
    const float* inp = in + (long)(b * IC + ic) * inS * inS2;
#pragma unroll
    for (int kd = 0; kd < 3; ++kd) {
      const int id = od * 2 - 1 + kd;
      if ((unsigned)id >= (unsigned)inS) continue;
#pragma unroll
      for (int kh = 0; kh < 3; ++kh) {
        const int ih = oh * 2 - 1 + kh;
        if ((unsigned)ih >= (unsigned)inS) continue;
#pragma unroll
        for (int kw = 0; kw < 3; ++kw) {
          const int iw = ow * 2 - 1 + kw;
          if ((unsigned)iw >= (unsigned)inS) continue;
          const float v = inp[id * inS2 + ih * inS + iw];
          const int tap = kd * 9 + kh * 3 + kw;
#pragma unroll
          for (int oc = 0; oc < OC; ++oc)
            s[oc] += v * w[(oc * IC + ic) * 27 + tap];  // uniform -> s_load
        }
      }
    }
  }
#pragma unroll
  for (int oc = 0; oc < OC; ++oc)
    out[(long)(b * OC + oc) * n + sp] = fmaxf(s[oc], 0.f);
}

// ---------------------------------------------------------------------------
// Refine conv: channel-concat of two tensors, zero-pad 1, 3x3x3, LeakyReLU(0.2)
// All OC per thread.  grid = (spatial/256, 1, B)
// ---------------------------------------------------------------------------
template <int ICA, int ICB, int OC>
__global__ void refine_conv_kernel(const float* __restrict__ inA,
                                   const float* __restrict__ inB,
                                   const float* __restrict__ w,
                                   const float* __restrict__ bias,
                                   float* __restrict__ out, int S) {
  const int b = blockIdx.z;
  const int S2 = S * S, n = S * S2;
  const int sp = blockIdx.x * blockDim.x + threadIdx.x;
  if (sp >= n) return;
  const int ow = sp % S;
  int t = sp / S;
  const int oh = t % S, od = t / S;
  const int IC = ICA + ICB;
  float s[OC];
#pragma unroll
  for (int oc = 0; oc < OC; ++oc) s[oc] = bias[oc];
  for (int ic = 0; ic < IC; ++ic) {
    const float* inp = (ic < ICA) ? inA + (long)(b * ICA + ic) * n
                                  : inB + (long)(b * ICB + (ic - ICA)) * n;
#pragma unroll
    for (int kd = 0; kd < 3; ++kd) {
      const int id = od - 1 + kd;
      if ((unsigned)id >= (unsigned)S) continue;
#pragma unroll
      for (int kh = 0; kh < 3; ++kh) {
        const int ih = oh - 1 + kh;
        if ((unsigned)ih >= (unsigned)S) continue;
#pragma unroll
        for (int kw = 0; kw < 3; ++kw) {
          const int iw = ow - 1 + kw;
          if ((unsigned)iw >= (unsigned)S) continue;
          const float v = inp[id * S2 + ih * S + iw];
          const int tap = kd * 9 + kh * 3 + kw;
#pragma unroll
          for (int oc = 0; oc < OC; ++oc)
            s[oc] += v * w[(oc * IC + ic) * 27 + tap];
        }
      }
    }
  }
#pragma unroll
  for (int oc = 0; oc < OC; ++oc)
    out[(long)(b * OC + oc) * n + sp] = (s[oc] > 0.f) ? s[oc] : 0.2f * s[oc];
}

// ---------------------------------------------------------------------------
// Residual-block conv: replicate-pad 1, 3x3x3.  All OC per thread.
//   MODE 0: out = leaky(conv(in))
//   MODE 1: out = leaky(conv(in) + res)
// ---------------------------------------------------------------------------
template <int C, int MODE>
__global__ void res_conv_kernel(const float* __restrict__ in,
                                const float* __restrict__ w,
                                const float* __restrict__ bias,
                                const float* __restrict__ res,
                                float* __restrict__ out, int S) {
  const int b = blockIdx.z;
  const int S2 = S * S, n = S * S2;
  const int sp = blockIdx.x * blockDim.x + threadIdx.x;
  if (sp >= n) return;
  const int ow = sp % S;
  int t = sp / S;
  const int oh = t % S, od = t / S;
  float s[C];
#pragma unroll
  for (int oc = 0; oc < C; ++oc) s[oc] = bias[oc];
  for (int ic = 0; ic < C; ++ic) {
    const float* inp = in + (long)(b * C + ic) * n;
#pragma unroll
    for (int kd = 0; kd < 3; ++kd) {
      const int id = min(max(od - 1 + kd, 0), S - 1);
#pragma unroll
      for (int kh = 0; kh < 3; ++kh) {
        const int ih = min(max(oh - 1 + kh, 0), S - 1);
#pragma unroll
        for (int kw = 0; kw < 3; ++kw) {
          const int iw = min(max(ow - 1 + kw, 0), S - 1);
          const float v = inp[id * S2 + ih * S + iw];
          const int tap = kd * 9 + kh * 3 + kw;
#pragma unroll
          for (int oc = 0; oc < C; ++oc)
            s[oc] += v * w[(oc * C + ic) * 27 + tap];
        }
      }
    }
  }
#pragma unroll
  for (int oc = 0; oc < C; ++oc) {
    float v = s[oc];
    if (MODE == 1) v += res[(long)(b * C + oc) * n + sp];
    out[(long)(b * C + oc) * n + sp] = (v > 0.f) ? v : 0.2f * v;
  }
}

// ---------------------------------------------------------------------------
// Transposed conv: kernel 6, stride 2, pad 2, no bias, ReLU.  All OC/thread.
// w layout: [IC][OC][6][6][6].  od = id*2 - 2 + kd -> id = (od+2-kd)/2,
// kd == od (mod 2), 3 taps/axis.
// ---------------------------------------------------------------------------
template <int IC, int OC>
__global__ void upconv_kernel(const float* __restrict__ in,
                              const float* __restrict__ w,
                              float* __restrict__ out, int S) {
  const int b = blockIdx.z;
  const int OS = S * 2, OS2 = OS * OS;
  const int n = OS * OS2;
  const int sp = blockIdx.x * blockDim.x + threadIdx.x;
  if (sp >= n) return;
  const int ow = sp % OS;
  int t = sp / OS;
  const int oh = t % OS, od = t / OS;
  const int S2 = S * S;
  const int pd = od & 1, ph = oh & 1, pw = ow & 1;
  float s[OC];
#pragma unroll
  for (int oc = 0; oc < OC; ++oc) s[oc] = 0.f;
  for (int ic = 0; ic < IC; ++ic) {
    const float* inp = in + (long)(b * IC + ic) * S * S2;
#pragma unroll
    for (int td = 0; td < 3; ++td) {
      const int kd = pd + 2 * td;
      const int id = (od + 2 - kd) >> 1;
      if ((unsigned)id >= (unsigned)S) continue;
#pragma unroll
      for (int th = 0; th < 3; ++th) {
        const int kh = ph + 2 * th;
        const int ih = (oh + 2 - kh) >> 1;
        if ((unsigned)ih >= (unsigned)S) continue;
#pragma unroll
        for (int tw = 0; tw < 3; ++tw) {
          const int kw = pw + 2 * tw;
          const int iw = (ow + 2 - kw) >> 1;
          if ((unsigned)iw >= (unsigned)S) continue;
          const float v = inp[id * S2 + ih * S + iw];
          const int tap = kd * 36 + kh * 6 + kw;
#pragma unroll
          for (int oc = 0; oc < OC; ++oc)
            s[oc] += v * w[(ic * OC + oc) * 216 + tap];
        }
      }
    }
  }
#pragma unroll
  for (int oc = 0; oc < OC; ++oc)
    out[(long)(b * OC + oc) * n + sp] = fmaxf(s[oc], 0.f);
}

// ---------------------------------------------------------------------------
// Window attention, one 4x4x4 window (64 tokens, C=4, 4 heads, head_dim=1)
// per 64-thread block (2 waves).  QKV projection runs on
// V_WMMA_F32_16X16X4_F32 (K=4 == channel count, exact tile fit).
// ---------------------------------------------------------------------------
__launch_bounds__(64) __global__
void win_attn_kernel(const float* __restrict__ enc2,
                     const float* __restrict__ rpb,
                     const float* __restrict__ qkv_w,
                     const float* __restrict__ qkv_b,
                     const float* __restrict__ proj_w,
                     const float* __restrict__ proj_b,
                     float* __restrict__ base) {
  __shared__ float xs[64][5];     // tokens x C (padded row to dodge conflicts)
  __shared__ float qkvs[12][65];  // j x token (padded)

  const int t = threadIdx.x;
  const int wi = blockIdx.x;              // 2048 windows
  const int b = wi >> 9;
  const int Db = (wi >> 6) & 7, Hb = (wi >> 3) & 7, Wb = wi & 7;
  const int nd = t >> 4, nh = (t >> 2) & 3, nw = t & 3;
  const int d = Db * 4 + nd, hh = Hb * 4 + nh, ww = Wb * 4 + nw;
  const long spi = (long)d * 1024 + hh * 32 + ww;

#pragma unroll
  for (int c = 0; c < 4; ++c)
    xs[t][c] = enc2[(long)(b * 4 + c) * 32768 + spi];
  __syncthreads();

  // ---- QKV projection: D[16 tok][16 col] = A[16x4] * B[4x16] (12 cols used)
  {
    const int lane = t & 31;
    const int wave = t >> 5;
    const int col = lane & 15;
    const int kb = (lane < 16) ? 0 : 2;  // K-half held by this lane group
    v2f bm;
    bm.x = (col < 12) ? qkv_w[col * 4 + kb] : 0.f;      // B[kb][col]
    bm.y = (col < 12) ? qkv_w[col * 4 + kb + 1] : 0.f;  // B[kb+1][col]
    const float cb = (col < 12) ? qkv_b[col] : 0.f;
#pragma unroll
    for (int i = 0; i < 2; ++i) {
      const int tile = wave * 2 + i;  // 16-token tile
      v2f am;
      am.x = xs[tile * 16 + col][kb];
      am.y = xs[tile * 16 + col][kb + 1];
      v8f acc = {};
      acc = __builtin_amdgcn_wmma_f32_16x16x4_f32(
          false, am, false, bm, (short)0, acc, false, false);
      const int rb = (lane < 16) ? 0 : 8;  // C/D row block per lane half
      if (col < 12) {
#pragma unroll
        for (int r = 0; r < 8; ++r)
          qkvs[col][tile * 16 + rb + r] = acc[r] + cb;
      }
    }
  }
  __syncthreads();

  // ---- per-token online softmax over 64 keys, 4 heads (head_dim = 1)
  float mx[4], sm[4], av[4], qh[4];
#pragma unroll
  for (int h = 0; h < 4; ++h) {
    mx[h] = -3.0e38f;
    sm[h] = 0.f;
    av[h] = 0.f;
    qh[h] = qkvs[h][t];  // scale = head_dim^-0.5 = 1
  }
  for (int m = 0; m < 64; ++m) {
    const int md = m >> 4, mh = (m >> 2) & 3, mw = m & 3;
    const int idx = (nd - md + 3) * 49 + (nh - mh + 3) * 7 + (nw - mw + 3);
    const float4 b4 = *reinterpret_cast<const float4*>(rpb + idx * 4);
    const float bias[4] = {b4.x, b4.y, b4.z, b4.w};
#pragma unroll
    for (int h = 0; h < 4; ++h) {
      const float s = qh[h] * qkvs[4 + h][m] + bias[h];
      const float nm = fmaxf(mx[h], s);
      const float corr = __expf(mx[h] - nm);
      const float e = __expf(s - nm);
      sm[h] = sm[h] * corr + e;
      av[h] = av[h] * corr + e * qkvs[8 + h][m];
      mx[h] = nm;
    }
  }
  float o[4];
#pragma unroll
  for (int h = 0; h < 4; ++h) o[h] = av[h] / sm[h];

  // ---- output projection (4x4) + scatter back to (b,c,d,h,w)
#pragma unroll
  for (int c = 0; c < 4; ++c) {
    float y = proj_b[c];
#pragma unroll
    for (int h = 0; h < 4; ++h) y += o[h] * proj_w[c * 4 + h];
    base[(long)(b * 4 + c) * 32768 + spi] = y;
  }
}

// ---------------------------------------------------------------------------
extern "C" void kernel_launch(void* const* d_in, const int* in_sizes, int n_in,
                              void* d_out, int out_size, void* d_ws,
                              size_t ws_size, hipStream_t stream) {
  (void)in_sizes; (void)n_in; (void)out_size; (void)ws_size;
  const float* x      = (const float*)d_in[0];
  const float* enc1_w = (const float*)d_in[1];
  const float* enc1_b = (const float*)d_in[2];
  const float* enc2_w = (const float*)d_in[3];
  const float* enc2_b = (const float*)d_in[4];
  const float* rpb    = (const float*)d_in[5];
  const float* qkv_w  = (const float*)d_in[6];
  const float* qkv_b  = (const float*)d_in[7];
  const float* proj_w = (const float*)d_in[8];
  const float* proj_b = (const float*)d_in[9];
  const float* d2_rw  = (const float*)d_in[10];
  const float* d2_rb  = (const float*)d_in[11];
  const float* d2_w1  = (const float*)d_in[12];
  const float* d2_b1  = (const float*)d_in[13];
  const float* d2_w2  = (const float*)d_in[14];
  const float* d2_b2  = (const float*)d_in[15];
  const float* d2_up  = (const float*)d_in[16];
  const float* d1_rw  = (const float*)d_in[17];
  const float* d1_rb  = (const float*)d_in[18];
  const float* d1_w1  = (const float*)d_in[19];
  const float* d1_b1  = (const float*)d_in[20];
  const float* d1_w2  = (const float*)d_in[21];
  const float* d1_b2  = (const float*)d_in[22];
  const float* d1_up  = (const float*)d_in[23];
  float* out = (float*)d_out;

  float* ws   = (float*)d_ws;
  float* enc1 = ws;               // (4,2,64^3)  = 2,097,152
  float* enc2 = enc1 + 2097152;   // (4,4,32^3)  =   524,288
  float* base = enc2 + 524288;    // (4,4,32^3)
  float* h2   = base + 524288;    // (4,4,32^3)
  float* r2a  = h2 + 524288;      // (4,4,32^3)
  float* r2   = r2a + 524288;     // (4,4,32^3)
  float* dec2 = r2 + 524288;      // (4,2,64^3)  = 2,097,152
  float* h1   = dec2 + 2097152;   // (4,2,64^3)
  float* r1a  = h1 + 2097152;     // (4,2,64^3)
  float* r1   = r1a + 2097152;    // (4,2,64^3)   total 13.1M floats (~52 MB)

  const dim3 blk(256);
  // encoders (all OC per thread)
  enc_conv_kernel<1, 2><<<dim3(1024, 1, 4), blk, 0, stream>>>(x, enc1_w, enc1_b, enc1, 128, 64);
  enc_conv_kernel<2, 4><<<dim3(128, 1, 4), blk, 0, stream>>>(enc1, enc2_w, enc2_b, enc2, 64, 32);
  // window attention (WMMA qkv projection)
  win_attn_kernel<<<dim3(2048), dim3(64), 0, stream>>>(enc2, rpb, qkv_w, qkv_b, proj_w, proj_b, base);
  // decoder stage 2 (32^3 -> 64^3)
  refine_conv_kernel<4, 4, 4><<<dim3(128, 1, 4), blk, 0, stream>>>(base, enc2, d2_rw, d2_rb, h2, 32);
  res_conv_kernel<4, 0><<<dim3(128, 1, 4), blk, 0, stream>>>(h2, d2_w1, d2_b1, nullptr, r2a, 32);
  res_conv_kernel<4, 1><<<dim3(128, 1, 4), blk, 0, stream>>>(r2a, d2_w2, d2_b2, h2, r2, 32);
  upconv_kernel<4, 2><<<dim3(1024, 1, 4), blk, 0, stream>>>(r2, d2_up, dec2, 32);
  // decoder stage 1 (64^3 -> 128^3)
  refine_conv_kernel<2, 2, 2><<<dim3(1024, 1, 4), blk, 0, stream>>>(dec2, enc1, d1_rw, d1_rb, h1, 64);
  res_conv_kernel<2, 0><<<dim3(1024, 1, 4), blk, 0, stream>>>(h1, d1_w1, d1_b1, nullptr, r1a, 64);
  res_conv_kernel<2, 1><<<dim3(1024, 1, 4), blk, 0, stream>>>(r1a, d1_w2, d1_b2, h1, r1, 64);
  upconv_kernel<2, 1><<<dim3(8192, 1, 4), blk, 0, stream>>>(r1, d1_up, out, 64);
}